// PointSampler_76459007804124
// MI455X (gfx1250) — compile-verified
//
#include <hip/hip_runtime.h>
#include <math.h>

#define NNODES 50000
#define NEDGES 1600000
#define OUTC   64

typedef __attribute__((ext_vector_type(2))) float v2f;
typedef __attribute__((ext_vector_type(8))) float v8f;

// Order-preserving encode: max over encoded uints == max over floats.
__device__ __forceinline__ unsigned enc_f32(float f) {
  unsigned u = __float_as_uint(f);
  return (u & 0x80000000u) ? ~u : (u | 0x80000000u);
}

// ---- Layer 0 theta: ht = x @ Tw0  (K=3, too small for WMMA) ----------------
__global__ __launch_bounds__(256) void devconv_theta3(
    const float* __restrict__ x, const float* __restrict__ tw,
    float* __restrict__ ht) {
  int idx = blockIdx.x * blockDim.x + threadIdx.x;
  if (idx >= NNODES * OUTC) return;
  int node = idx >> 6;
  int c    = idx & 63;
  float x0 = x[node * 3 + 0], x1 = x[node * 3 + 1], x2 = x[node * 3 + 2];
  ht[idx] = fmaf(x0, tw[c], fmaf(x1, tw[64 + c], x2 * tw[128 + c]));
}

// ---- init scatter-max buffer to encoded(-inf) ------------------------------
__global__ __launch_bounds__(256) void devconv_init_max(unsigned* __restrict__ m) {
  int idx = blockIdx.x * blockDim.x + threadIdx.x;
  if (idx < NNODES * OUTC) m[idx] = 0x007FFFFFu;  // enc_f32(-inf)
}

// ---- edge scatter: m[dst][c] = max(m[dst][c], ht[src][c]) ------------------
// 16 threads per edge, each handling a float4 slice of the 64-ch row:
// fully coalesced 256B row reads; native global_atomic_max_u32 (L2-resident).
__global__ __launch_bounds__(256) void devconv_scatter_max(
    const float4* __restrict__ ht4, const int* __restrict__ src,
    const int* __restrict__ dst, unsigned* __restrict__ m) {
  int idx = blockIdx.x * blockDim.x + threadIdx.x;
  if (idx >= NEDGES * 16) return;
  int e = idx >> 4;
  int q = idx & 15;
  int s = __ldg(src + e);
  int d = __ldg(dst + e);
  float4 v = ht4[(size_t)s * 16 + q];
  unsigned* mp = m + (size_t)d * 64 + q * 4;
  atomicMax(mp + 0, enc_f32(v.x));
  atomicMax(mp + 1, enc_f32(v.y));
  atomicMax(mp + 2, enc_f32(v.z));
  atomicMax(mp + 3, enc_f32(v.w));
}

// ---- combine (in place): agg = (m==-inf) ? 0 : m - ht[dst] + tb ------------
__global__ __launch_bounds__(256) void devconv_combine(
    unsigned* __restrict__ m, const float* __restrict__ ht,
    const float* __restrict__ tb) {
  int idx = blockIdx.x * blockDim.x + threadIdx.x;
  if (idx >= NNODES * OUTC) return;
  unsigned u = m[idx];
  float v = (u & 0x80000000u) ? __uint_as_float(u ^ 0x80000000u)
                              : __uint_as_float(~u);
  int c = idx & 63;
  float r = (v == -INFINITY) ? 0.0f : (v - ht[idx] + tb[c]);
  ((float*)m)[idx] = r;
}

// ---- WMMA GEMM: Y[N x 64] = X[N x 64] @ W[64 x 64] (+bias, relu) -----------
// One wave per 16x16 output tile; K=64 fully unrolled as 16x v_wmma_f32_16x16x4_f32.
// A lane layout: M = lane&15, K-pair offset = 2*(lane>>4). B symmetric. D per ISA.
template <bool BIASRELU>
__global__ __launch_bounds__(256) void gemm_n64_k64_wmma(
    const float* __restrict__ X, const float* __restrict__ W,
    const float* __restrict__ bias, float* __restrict__ Y) {
  const int tid  = threadIdx.x;
  const int wave = tid >> 5;
  const int lane = tid & 31;
  const int rowTile = blockIdx.x * 2 + (wave >> 2);   // 2 row-tiles per block
  const int m0 = rowTile * 16;
  if (m0 >= NNODES) return;                            // wave-uniform exit
  const int n0   = (wave & 3) * 16;
  const int M    = lane & 15;
  const int hi   = lane >> 4;
  const int koff = hi * 2;

  const float* xr = X + (size_t)(m0 + M) * 64 + koff;  // A row for this lane
  const float* wr = W + (size_t)koff * 64 + n0 + M;    // B col for this lane

  v8f acc = {0.f, 0.f, 0.f, 0.f, 0.f, 0.f, 0.f, 0.f};
#pragma unroll
  for (int k = 0; k < 64; k += 4) {
    v2f a, b;
    a.x = xr[k];
    a.y = xr[k + 1];
    b.x = wr[(size_t)k * 64];
    b.y = wr[(size_t)k * 64 + 64];
    acc = __builtin_amdgcn_wmma_f32_16x16x4_f32(
        /*neg_a=*/false, a, /*neg_b=*/false, b,
        /*c_mod=*/(short)0, acc, /*reuse_a=*/false, /*reuse_b=*/false);
  }

  float bv = BIASRELU ? bias[n0 + M] : 0.0f;
#pragma unroll
  for (int r = 0; r < 8; ++r) {
    float v = acc[r];
    if (BIASRELU) v = fmaxf(v + bv, 0.0f);
    Y[(size_t)(m0 + r + 8 * hi) * 64 + n0 + M] = v;
  }
}

// ---- final: sigmoid(h @ out_w + out_b), one wave per node ------------------
__global__ __launch_bounds__(256) void devconv_score(
    const float* __restrict__ h, const float* __restrict__ ow,
    const float* __restrict__ ob, float* __restrict__ out) {
  int g    = blockIdx.x * blockDim.x + threadIdx.x;
  int node = g >> 5;
  int lane = g & 31;
  if (node >= NNODES) return;
  const float* row = h + (size_t)node * 64;
  float s = fmaf(row[lane], ow[lane], row[lane + 32] * ow[lane + 32]);
#pragma unroll
  for (int off = 16; off > 0; off >>= 1) s += __shfl_down(s, off, 32);
  if (lane == 0) {
    float z = s + ob[0];
    out[node] = 1.0f / (1.0f + __expf(-z));
  }
}

extern "C" void kernel_launch(void* const* d_in, const int* in_sizes, int n_in,
                              void* d_out, int out_size, void* d_ws, size_t ws_size,
                              hipStream_t stream) {
  (void)in_sizes; (void)n_in; (void)out_size; (void)ws_size;
  const float* x  = (const float*)d_in[0];
  const int*   ei = (const int*)d_in[1];
  const int*   src = ei;            // edge_index[0]
  const int*   dst = ei + NEDGES;   // edge_index[1]
  const float* tw[3] = {(const float*)d_in[2],  (const float*)d_in[3],  (const float*)d_in[4]};
  const float* tb[3] = {(const float*)d_in[5],  (const float*)d_in[6],  (const float*)d_in[7]};
  const float* pw[3] = {(const float*)d_in[8],  (const float*)d_in[9],  (const float*)d_in[10]};
  const float* pb[3] = {(const float*)d_in[11], (const float*)d_in[12], (const float*)d_in[13]};
  const float* ow = (const float*)d_in[14];
  const float* ob = (const float*)d_in[15];

  float*    h  = (float*)d_ws;                       // current features [N,64]
  float*    ht = h + (size_t)NNODES * OUTC;          // h @ Tw           [N,64]
  unsigned* m  = (unsigned*)(ht + (size_t)NNODES * OUTC);  // scatter-max / agg

  const int nodeElems = NNODES * OUTC;
  dim3 blk(256);
  dim3 gNode((nodeElems + 255) / 256);
  dim3 gScat((NEDGES * 16 + 255) / 256);
  const int rowTiles = (NNODES + 15) / 16;           // 3125 (exact)
  dim3 gGemm((rowTiles + 1) / 2);
  dim3 gScore((NNODES * 32 + 255) / 256);

  for (int l = 0; l < 3; ++l) {
    if (l == 0)
      devconv_theta3<<<gNode, blk, 0, stream>>>(x, tw[0], ht);
    else
      gemm_n64_k64_wmma<false><<<gGemm, blk, 0, stream>>>(h, tw[l], nullptr, ht);
    devconv_init_max<<<gNode, blk, 0, stream>>>(m);
    devconv_scatter_max<<<gScat, blk, 0, stream>>>((const float4*)ht, src, dst, m);
    devconv_combine<<<gNode, blk, 0, stream>>>(m, ht, tb[l]);
    gemm_n64_k64_wmma<true><<<gGemm, blk, 0, stream>>>((const float*)m, pw[l], pb[l], h);
  }
  devconv_score<<<gScore, blk, 0, stream>>>(h, ow, ob, (float*)d_out);
}